// MinMaxDiffSetFeat_2233382994386
// MI455X (gfx1250) — compile-verified
//
#include <hip/hip_runtime.h>

// CDNA5 / gfx1250, wave32. FP32 WMMA path (V_WMMA_F32_16X16X4_F32).

typedef __attribute__((ext_vector_type(2))) float v2f;
typedef __attribute__((ext_vector_type(8))) float v8f;

#define NROWS   500000
#define DF      64            // D_IN == D_OUT
#define INMLP   192           // 3 * D_IN
#define NGROUPS 50000
#define TILES_PER_BLOCK 8
#define HC_STRIDE 196         // 192 padded: 196 % 64 == 4 -> conflict-free A reads
#define H1_STRIDE 68          // 64 padded:   68 % 64 == 4

// ---------------- Kernel 1: per-group min/max + segment-id scatter ----------
__global__ void seg_minmax_kernel(const float* __restrict__ x,
                                  const int*   __restrict__ csr,
                                  float* __restrict__ smin,
                                  float* __restrict__ smax,
                                  int*   __restrict__ seg) {
    const int g = blockIdx.x;          // group
    const int t = threadIdx.x;         // 0..63 = feature column
    const int p0 = csr[g];
    const int p1 = csr[g + 1];
    float mn = INFINITY, mx = -INFINITY;
    for (int r = p0; r < p1; ++r) {
        float v = x[(long)r * DF + t]; // 64 lanes read one 256B row: coalesced
        mn = fminf(mn, v);
        mx = fmaxf(mx, v);
    }
    smin[(long)g * DF + t] = mn;
    smax[(long)g * DF + t] = mx;
    for (int r = p0 + t; r < p1; r += 64) seg[r] = g;  // row -> group id
}

// ---------------- Kernel 2: fused concat + 2-layer MLP via f32 WMMA ---------
// Block = 128 threads (4 waves). Wave j owns output columns [16j, 16j+16) of
// both GEMMs; its B-fragments (weight slices) live in registers for the whole
// block. Each block processes TILES_PER_BLOCK row-tiles of 16.
__global__ __launch_bounds__(128) void fused_mlp_kernel(
        const float* __restrict__ x,
        const int*   __restrict__ seg,
        const float* __restrict__ smin,
        const float* __restrict__ smax,
        const float* __restrict__ w1,   // [192][64] row-major
        const float* __restrict__ w2,   // [64][64]  row-major
        float* __restrict__ out) {      // [N][64]
    __shared__ __align__(16) float hcat[16 * HC_STRIDE]; // also reused as out-staging
    __shared__ __align__(16) float h1[16 * H1_STRIDE];

    const int tid   = threadIdx.x;     // 0..127
    const int lane  = tid & 31;
    const int wave  = tid >> 5;        // n-tile index j
    const int col16 = lane & 15;       // N (B/C) or M (A) position
    const int koff  = (lane < 16) ? 0 : 2;  // A/B K-offset per half-wave

    // ---- preload this wave's weight slices as WMMA B fragments (registers)
    // B 4x16 f32 layout (mirror of A 16x4): VGPR0 = rows K=0 (lanes 0-15) /
    // K=2 (lanes 16-31); VGPR1 = rows K=1 / K=3.
    v2f w1f[48];
    v2f w2f[16];
    {
        const float* p = w1 + wave * 16 + col16;
        #pragma unroll
        for (int k = 0; k < 48; ++k) {
            w1f[k].x = p[(4 * k + koff) * DF];
            w1f[k].y = p[(4 * k + koff + 1) * DF];
        }
        const float* q = w2 + wave * 16 + col16;
        #pragma unroll
        for (int k = 0; k < 16; ++k) {
            w2f[k].x = q[(4 * k + koff) * DF];
            w2f[k].y = q[(4 * k + koff + 1) * DF];
        }
    }

    for (int t = 0; t < TILES_PER_BLOCK; ++t) {
        const long tile = (long)blockIdx.x * TILES_PER_BLOCK + t;
        const long row0 = tile * 16;
        if (row0 >= NROWS) break;      // uniform across block

        // ---- build concat tile [16 x 192] in LDS: x | x-min | x-max
        #pragma unroll
        for (int i = 0; i < 8; ++i) {
            int e = tid + i * 128;     // 0..1023 -> coalesced over x rows
            int r = e >> 6;
            int c = e & 63;
            long gr = row0 + r;
            float xv = x[gr * DF + c];
            int   s  = seg[gr];
            float mn = smin[(long)s * DF + c];
            float mx = smax[(long)s * DF + c];
            hcat[r * HC_STRIDE + c]        = xv;
            hcat[r * HC_STRIDE + 64 + c]  = xv - mn;
            hcat[r * HC_STRIDE + 128 + c] = xv - mx;
        }
        __syncthreads();

        // ---- layer 1: C[16x16] = hcat[16x192] x W1[:,16j:16j+16]
        v8f c1 = {0.f, 0.f, 0.f, 0.f, 0.f, 0.f, 0.f, 0.f};
        {
            const float* ha = &hcat[col16 * HC_STRIDE];  // A row for this lane
            #pragma unroll
            for (int k = 0; k < 48; ++k) {
                v2f a;
                a.x = ha[4 * k + koff];        // adjacent dwords -> ds_load_b64
                a.y = ha[4 * k + koff + 1];
                c1 = __builtin_amdgcn_wmma_f32_16x16x4_f32(
                        false, a, false, w1f[k], (short)0, c1, false, false);
            }
        }
        // LeakyReLU(0.2) + re-stripe through LDS for layer-2 A fragments
        #pragma unroll
        for (int m = 0; m < 8; ++m) {
            float v = c1[m];
            v = (v >= 0.f) ? v : 0.2f * v;
            int rowm = m + ((lane < 16) ? 0 : 8);   // C layout: VGPR m -> row m / m+8
            h1[rowm * H1_STRIDE + wave * 16 + col16] = v;
        }
        __syncthreads();

        // ---- layer 2: C[16x16] = h1[16x64] x W2[:,16j:16j+16]
        v8f c2 = {0.f, 0.f, 0.f, 0.f, 0.f, 0.f, 0.f, 0.f};
        {
            const float* ha = &h1[col16 * H1_STRIDE];
            #pragma unroll
            for (int k = 0; k < 16; ++k) {
                v2f a;
                a.x = ha[4 * k + koff];
                a.y = ha[4 * k + koff + 1];
                c2 = __builtin_amdgcn_wmma_f32_16x16x4_f32(
                        false, a, false, w2f[k], (short)0, c2, false, false);
            }
        }
        // LeakyReLU + stage result tile [16x64] into hcat region (done with it)
        #pragma unroll
        for (int m = 0; m < 8; ++m) {
            float v = c2[m];
            v = (v >= 0.f) ? v : 0.2f * v;
            int rowm = m + ((lane < 16) ? 0 : 8);
            hcat[rowm * DF + wave * 16 + col16] = v;   // unpadded 16x64 staging
        }
        __syncthreads();

        // ---- coalesced float4 store of the tile: 256 vec4 / 128 threads
        #pragma unroll
        for (int i = 0; i < 2; ++i) {
            int v4 = tid + i * 128;                    // 0..255
            float4 val = ((const float4*)hcat)[v4];
            ((float4*)(out + row0 * DF))[v4] = val;
        }
        __syncthreads();   // protect staging buffer before next tile's fill
    }
}

extern "C" void kernel_launch(void* const* d_in, const int* in_sizes, int n_in,
                              void* d_out, int out_size, void* d_ws, size_t ws_size,
                              hipStream_t stream) {
    const float* x   = (const float*)d_in[0];   // [500000*64]
    const int*   csr = (const int*)d_in[1];     // [50001]
    const float* w1  = (const float*)d_in[2];   // [192*64]
    const float* w2  = (const float*)d_in[3];   // [64*64]
    float* out = (float*)d_out;                 // [500000*64]

    // workspace layout: seg_min[G*64] f32 | seg_max[G*64] f32 | seg[N] i32
    float* smin = (float*)d_ws;
    float* smax = smin + (size_t)NGROUPS * DF;
    int*   seg  = (int*)(smax + (size_t)NGROUPS * DF);

    seg_minmax_kernel<<<NGROUPS, 64, 0, stream>>>(x, csr, smin, smax, seg);

    const int n_tiles = NROWS / 16;                       // 31250 exact
    const int blocks  = (n_tiles + TILES_PER_BLOCK - 1) / TILES_PER_BLOCK;
    fused_mlp_kernel<<<blocks, 128, 0, stream>>>(x, seg, smin, smax, w1, w2, out);
}